// DeltaNetLayer_90237262889420
// MI455X (gfx1250) — compile-verified
//
#include <hip/hip_runtime.h>
#include <hip/hip_bf16.h>
#include <math.h>

#define HID   2048
#define INTER 8192
#define QKV   8192
#define TV    4096
#define NVH   32
#define KD    128
#define VD    128
#define NKH   16
#define CK    4

typedef float v2f __attribute__((ext_vector_type(2)));
typedef float v4f __attribute__((ext_vector_type(4)));
typedef float v8f __attribute__((ext_vector_type(8)));

// ---------------------------------------------------------------------------
// Block reduction (wave32), blockDim.x <= 256
// ---------------------------------------------------------------------------
__device__ __forceinline__ float block_reduce_sum(float v) {
    __shared__ float sbuf[8];
    const int lane = threadIdx.x & 31;
    const int w    = threadIdx.x >> 5;
#pragma unroll
    for (int off = 16; off > 0; off >>= 1) v += __shfl_down(v, off, 32);
    if (lane == 0) sbuf[w] = v;
    __syncthreads();
    float r = 0.f;
    if (threadIdx.x < (blockDim.x >> 5)) r = sbuf[threadIdx.x];
    if (w == 0) {
#pragma unroll
        for (int off = 4; off > 0; off >>= 1) r += __shfl_down(r, off, 32);
        if (lane == 0) sbuf[0] = r;
    }
    __syncthreads();
    float res = sbuf[0];
    __syncthreads();
    return res;
}

__device__ __forceinline__ float silu_f(float x) { return x / (1.f + expf(-x)); }

// ---------------------------------------------------------------------------
// RMSNorm: out = x * rsqrt(mean(x^2)+eps) * (1+w)   (single block)
// ---------------------------------------------------------------------------
__global__ void __launch_bounds__(256) rmsnorm_kernel(
    const float* __restrict__ x, const float* __restrict__ w,
    float* __restrict__ out, int n)
{
    float ss = 0.f;
    for (int i = threadIdx.x; i < n; i += blockDim.x) { float v = x[i]; ss += v * v; }
    float tot = block_reduce_sum(ss);
    float inv = rsqrtf(tot / (float)n + 1e-6f);
    for (int i = threadIdx.x; i < n; i += blockDim.x)
        out[i] = x[i] * inv * (1.f + w[i]);
}

// ---------------------------------------------------------------------------
// fp32 GEMV via V_WMMA_F32_16X16X4_F32. One wave = 16 output rows.
// K-step of 8: one NT b128 weight load per lane (lo lanes: logical k..k+3,
// hi lanes: k+4..k+7) feeds TWO WMMAs; the K-index permutation is legal since
// B (broadcast x chunk, from one ds b128) uses the matching logical k.
// D[m,n] holds dot(W[row0+m,:],x) in every column; column N=0 (lanes 0/16)
// is written out. Split-K partials via blockIdx.y (dst = part + split*R).
// ---------------------------------------------------------------------------
__global__ void __launch_bounds__(256) gemv_wmma_kernel(
    const float* __restrict__ W, const float* __restrict__ x,
    float* __restrict__ dst, int R, int K, int Kchunk)
{
    __shared__ __align__(16) float xs[2048];       // Kchunk <= 2048
    const int split = blockIdx.y;
    const int k0    = split * Kchunk;

    for (int i = threadIdx.x * 4; i < Kchunk; i += blockDim.x * 4)
        *(v4f*)&xs[i] = *(const v4f*)&x[k0 + i];
    __syncthreads();

    const int lane = threadIdx.x & 31;
    const int waveGlobal = blockIdx.x * (blockDim.x >> 5) + (threadIdx.x >> 5);
    const int row0 = waveGlobal << 4;
    if (row0 >= R) return;                 // wave-uniform: EXEC all-1s for WMMA

    const int m   = lane & 15;
    const int off = (lane >> 4) << 2;      // lo lanes: +0, hi lanes: +4 floats
    const float* wrow = W + (size_t)(row0 + m) * (size_t)K + (size_t)(k0 + off);

    v8f c = {0.f, 0.f, 0.f, 0.f, 0.f, 0.f, 0.f, 0.f};
    for (int k = 0; k < Kchunk; k += 32) {
        __builtin_prefetch(wrow + k + 1024, 0, 0);   // ~4KB ahead in row stream
#pragma unroll
        for (int kk = 0; kk < 32; kk += 8) {
            v4f a4 = __builtin_nontemporal_load((const v4f*)(wrow + k + kk));
            v4f bx = *(const v4f*)&xs[k + kk + off];
            v2f a1 = {a4.x, a4.y}, a2 = {a4.z, a4.w};
            v2f b1 = {bx.x, bx.y}, b2 = {bx.z, bx.w};
            c = __builtin_amdgcn_wmma_f32_16x16x4_f32(
                    false, a1, false, b1, (short)0, c, false, false);
            c = __builtin_amdgcn_wmma_f32_16x16x4_f32(
                    false, a2, false, b2, (short)0, c, false, false);
        }
    }

    if (m == 0) {                          // lanes 0 and 16 hold column N=0
        const int base = (lane >> 4) << 3; // rows 0..7 or 8..15
        float* o = dst + (size_t)split * (size_t)R + (size_t)(row0 + base);
        v4f lo4 = {c[0], c[1], c[2], c[3]};
        v4f hi4 = {c[4], c[5], c[6], c[7]};
        *(v4f*)(o)     = lo4;
        *(v4f*)(o + 4) = hi4;
    }
}

__global__ void reduce_part_kernel(const float* __restrict__ part,
                                   const float* __restrict__ resid,
                                   float* __restrict__ out, int R, int S)
{
    int i = blockIdx.x * blockDim.x + threadIdx.x;
    if (i >= R) return;
    float s = resid ? resid[i] : 0.f;
    for (int j = 0; j < S; ++j) s += part[(size_t)j * (size_t)R + i];
    out[i] = s;
}

// Fused: x1 = resid + sum(part) ; hn = rmsnorm(x1, w) with (1+w). Single block.
__global__ void __launch_bounds__(256) reduce_norm_kernel(
    const float* __restrict__ part, const float* __restrict__ resid,
    const float* __restrict__ w, float* __restrict__ x1,
    float* __restrict__ hn, int R, int S)
{
    float xv[8];                                   // R/blockDim <= 8
    float ss = 0.f;
    int idx = 0;
    for (int i = threadIdx.x; i < R; i += blockDim.x, ++idx) {
        float v = resid[i];
        for (int s = 0; s < S; ++s) v += part[(size_t)s * (size_t)R + i];
        x1[i] = v; xv[idx] = v; ss += v * v;
    }
    float tot = block_reduce_sum(ss);
    float inv = rsqrtf(tot / (float)R + 1e-6f);
    idx = 0;
    for (int i = threadIdx.x; i < R; i += blockDim.x, ++idx)
        hn[i] = xv[idx] * inv * (1.f + w[i]);
}

// Fused: act = silu(gate) * sum(part_up)
__global__ void reduce_swiglu_kernel(const float* __restrict__ part,
                                     const float* __restrict__ gate,
                                     float* __restrict__ act, int R, int S)
{
    int i = blockIdx.x * blockDim.x + threadIdx.x;
    if (i >= R) return;
    float u = 0.f;
    for (int j = 0; j < S; ++j) u += part[(size_t)j * (size_t)R + i];
    act[i] = silu_f(gate[i]) * u;
}

// ---------------------------------------------------------------------------
// Conv state shift + depthwise conv + silu. Writes new_conv_state to d_out.
// ---------------------------------------------------------------------------
__global__ void __launch_bounds__(256) conv_kernel(
    const float* __restrict__ cs, const float* __restrict__ proj,
    const float* __restrict__ cw, float* __restrict__ ncs,
    float* __restrict__ conv_out)
{
    int c = blockIdx.x * blockDim.x + threadIdx.x;
    if (c >= QKV) return;
    v4f s = *(const v4f*)(cs + (size_t)c * CK);
    v4f w = *(const v4f*)(cw + (size_t)c * CK);
    float s3 = proj[c];
    v4f n = {s.y, s.z, s.w, s3};
    *(v4f*)(ncs + (size_t)c * CK) = n;
    float acc = s.y * w.x + s.z * w.y + s.w * w.z + s3 * w.w;
    conv_out[c] = silu_f(acc);
}

// ---------------------------------------------------------------------------
// Fused q/k l2norm + beta/decay + delta-rule state update + gated readout.
// grid = NVH blocks, 256 threads: thread = (v, k-half). State access is
// v-major (coalesced); Sk / y pair-reduced through LDS.
// ---------------------------------------------------------------------------
__global__ void __launch_bounds__(256) ssm_fused_kernel(
    const float* __restrict__ conv_out, const float* __restrict__ proj,
    const float* __restrict__ A_log, const float* __restrict__ dt_bias,
    const float* __restrict__ S, const float* __restrict__ norm_w,
    float* __restrict__ S_out, float* __restrict__ y_gated)
{
    const int h    = blockIdx.x;
    const int v    = threadIdx.x & (VD - 1);
    const int half = threadIdx.x >> 7;             // 0 or 1
    const int kh   = h >> 1;                       // RATIO = 2

    __shared__ float kl[KD], ql[KD];
    __shared__ float sc[2];
    __shared__ float rbuf[256];

    float qv = conv_out[kh * KD + v];
    float kv = conv_out[NKH * KD + kh * KD + v];
    float qss = block_reduce_sum(half == 0 ? qv * qv : 0.f);
    float kss = block_reduce_sum(half == 0 ? kv * kv : 0.f);
    if (half == 0) {
        ql[v] = qv / fmaxf(sqrtf(qss), 1e-12f);
        kl[v] = kv / fmaxf(sqrtf(kss), 1e-12f);
    }
    if (threadIdx.x == 0) {
        float araw = proj[QKV + TV + h];
        float braw = proj[QKV + TV + NVH + h];
        float xin  = araw + dt_bias[h];
        float sp   = (xin > 20.f) ? xin : log1pf(expf(xin));
        sc[0] = 1.f / (1.f + expf(-braw));         // beta
        sc[1] = expf(-expf(A_log[h]) * sp);        // decay
    }
    __syncthreads();
    const float bta = sc[0], dec = sc[1];

    const float* Sh = S     + (size_t)h * KD * VD;
    float*       So = S_out + (size_t)h * KD * VD;
    const int kbeg = half * (KD / 2), kend = kbeg + KD / 2;

    float Skp = 0.f;
#pragma unroll 4
    for (int k = kbeg; k < kend; ++k) Skp += Sh[k * VD + v] * kl[k];
    rbuf[threadIdx.x] = Skp;
    __syncthreads();
    const float Sk    = rbuf[v] + rbuf[v + 128];
    const float delta = conv_out[2 * NKH * KD + h * VD + v] - Sk;
    __syncthreads();

    float yp = 0.f;
#pragma unroll 4
    for (int k = kbeg; k < kend; ++k) {
        float sn = dec * Sh[k * VD + v] + bta * kl[k] * delta;
        So[k * VD + v] = sn;
        yp += sn * ql[k];
    }
    rbuf[threadIdx.x] = yp;
    __syncthreads();
    const float y = rbuf[v] + rbuf[v + 128];
    float ss = block_reduce_sum(half == 0 ? y * y : 0.f);
    if (half == 0) {
        float ynorm = y * rsqrtf(ss / (float)VD + 1e-6f) * norm_w[v];
        float z = proj[QKV + h * VD + v];
        y_gated[h * VD + v] = ynorm * silu_f(z);
    }
}

// ---------------------------------------------------------------------------
static void launch_gemv(const float* W, const float* x, float* dst,
                        int R, int K, int S, hipStream_t stream)
{
    const int Kchunk = K / S;                 // multiple of 32, <= 2048
    const int waves  = R / 16;                // R is a multiple of 16 here
    const int blocks = (waves + 7) / 8;       // 8 waves (256 threads) per block
    dim3 grid(blocks, S);
    gemv_wmma_kernel<<<grid, 256, 0, stream>>>(W, x, dst, R, K, Kchunk);
}

extern "C" void kernel_launch(void* const* d_in, const int* in_sizes, int n_in,
                              void* d_out, int out_size, void* d_ws, size_t ws_size,
                              hipStream_t stream)
{
    const float* x          = (const float*)d_in[0];
    const float* conv_state = (const float*)d_in[1];
    const float* ssm_state  = (const float*)d_in[2];
    const float* in_ln_w    = (const float*)d_in[3];
    const float* in_proj_w  = (const float*)d_in[4];
    const float* conv_w     = (const float*)d_in[5];
    const float* A_log      = (const float*)d_in[6];
    const float* dt_bias    = (const float*)d_in[7];
    const float* norm_w     = (const float*)d_in[8];
    const float* out_proj_w = (const float*)d_in[9];
    const float* post_ln_w  = (const float*)d_in[10];
    const float* gate_w     = (const float*)d_in[11];
    const float* up_w       = (const float*)d_in[12];
    const float* down_w     = (const float*)d_in[13];

    float* out      = (float*)d_out;
    float* out_x    = out;                    // 2048
    float* out_conv = out + HID;              // 8192*4
    float* out_ssm  = out + HID + QKV * CK;   // 32*128*128

    float* ws       = (float*)d_ws;
    float* h        = ws;                     // 2048
    float* proj     = ws + 2048;              // 12352
    float* conv_out = ws + 14400;             // 8192
    float* y_gated  = ws + 22592;             // 4096
    float* x1       = ws + 26688;             // 2048
    float* hn       = ws + 28736;             // 2048
    float* gate_out = ws + 30784;             // 8192
    float* act      = ws + 38976;             // 8192
    float* part     = ws + 47168;             // 16384 (split-K partials)

    const int RPROJ = QKV + TV + 2 * NVH;     // 12352

    // 1) h = rmsnorm(x, in_ln_w)
    rmsnorm_kernel<<<1, 256, 0, stream>>>(x, in_ln_w, h, HID);

    // 2) proj = in_proj_w @ h  (S=1: GEMV writes proj directly, 772 waves)
    launch_gemv(in_proj_w, h, proj, RPROJ, HID, 1, stream);

    // 3) conv shift + silu; writes new_conv_state
    conv_kernel<<<QKV / 256, 256, 0, stream>>>(conv_state, proj, conv_w, out_conv, conv_out);

    // 4) fused l2norm/beta/decay + delta-rule update + gated readout
    ssm_fused_kernel<<<NVH, 256, 0, stream>>>(conv_out, proj, A_log, dt_bias,
                                              ssm_state, norm_w, out_ssm, y_gated);

    // 5) attn_out partials (split-K=4), then fused residual + post rmsnorm
    launch_gemv(out_proj_w, y_gated, part, HID, TV, 4, stream);
    reduce_norm_kernel<<<1, 256, 0, stream>>>(part, x, post_ln_w, x1, hn, HID, 4);

    // 6) gate / up GEMVs (split-K=2 -> 1024 waves each)
    launch_gemv(gate_w, hn, part, INTER, HID, 2, stream);
    reduce_part_kernel<<<INTER / 256, 256, 0, stream>>>(part, nullptr, gate_out, INTER, 2);
    launch_gemv(up_w, hn, part, INTER, HID, 2, stream);
    reduce_swiglu_kernel<<<INTER / 256, 256, 0, stream>>>(part, gate_out, act, INTER, 2);

    // 7) x_out = x1 + down_w @ act   (split-K=8 -> 1024 waves)
    launch_gemv(down_w, act, part, HID, INTER, 8, stream);
    reduce_part_kernel<<<HID / 256, 256, 0, stream>>>(part, x1, out_x, HID, 8);
}